// ImplicitVAE_33071248179563
// MI455X (gfx1250) — compile-verified
//
#include <hip/hip_runtime.h>
#include <hip/hip_bf16.h>
#include <math.h>
#include <stdint.h>

// ---------------- problem constants ----------------
#define HD 64
static constexpr int N_NODES = 50000;
static constexpr int N_EDGES = 800000;
static constexpr int N_GRAPH = 2500;
static constexpr float BN_EPS = 1e-5f;
static constexpr int NBLK = 512;   // grid for stat-producing kernels
static constexpr int XBLK = 256;   // grid for x scalar stats

typedef __attribute__((ext_vector_type(2))) float v2f;
typedef __attribute__((ext_vector_type(8))) float v8f;

__device__ __forceinline__ float softplus_f(float v) {
    return v > 30.f ? v : log1pf(__expf(v));
}

__device__ __forceinline__ v8f vzero8() {
    v8f c;
    #pragma unroll
    for (int i = 0; i < 8; i++) c[i] = 0.f;
    return c;
}

__device__ __forceinline__ v8f wmma_f32_k4(v2f a, v2f b, v8f c) {
    // D = A(16x4) * B(4x16) + C, fp32, wave32
    return __builtin_amdgcn_wmma_f32_16x16x4_f32(false, a, false, b, (short)0, c,
                                                 false, false);
}

// ---- CDNA5 async global->LDS copy (ASYNCcnt path) ----
// Generic pointers to LDS are {SHARED_BASE_HI32, lds_offset32}; low 32 bits are
// the wave-relative LDS byte address the instruction wants in VDST.
__device__ __forceinline__ void async_copy_b128(void* lds_ptr, const void* gptr) {
    uint32_t lds = (uint32_t)(uintptr_t)lds_ptr;
    uint64_t ga  = (uint64_t)(uintptr_t)gptr;
    asm volatile("global_load_async_to_lds_b128 %0, %1, off"
                 :: "v"(lds), "v"(ga) : "memory");
}
__device__ __forceinline__ void wait_async0() {
    asm volatile("s_wait_asynccnt 0x0" ::: "memory");
}
__device__ __forceinline__ void wait_ds0() {
    asm volatile("s_wait_dscnt 0x0" ::: "memory");
}

// A: LDS row-major [16][LDA]; B: LDS [K][LDB] (column offset pre-applied).
// Fragment layout per ISA: A lane l16 = row M, vgpr0/1 = K (+2 for upper half);
// B lanes = N cols, vgpr0/1 = K rows.
template <int K, int LDA, int LDB>
__device__ __forceinline__ v8f gemm_tile(const float* A, const float* B,
                                         int l16, int half, v8f c) {
    const float* arow = A + l16 * LDA + 2 * half;
    #pragma unroll
    for (int k0 = 0; k0 < K; k0 += 4) {
        v2f a, b;
        a.x = arow[k0];
        a.y = arow[k0 + 1];
        b.x = B[(k0 + 2 * half) * LDB + l16];
        b.y = B[(k0 + 2 * half + 1) * LDB + l16];
        c = wmma_f32_k4(a, b, c);
    }
    return c;
}

// ---------------- scalar x stats (for analytic BN of d_fc1) ----------------
__global__ void k_xstats_partial(const float* __restrict__ x, float* __restrict__ xpart) {
    __shared__ float ss[256], sq[256];
    float s = 0.f, q = 0.f;
    for (int i = blockIdx.x * blockDim.x + threadIdx.x; i < N_EDGES;
         i += gridDim.x * blockDim.x) {
        float v = x[i]; s += v; q += v * v;
    }
    ss[threadIdx.x] = s; sq[threadIdx.x] = q;
    __syncthreads();
    for (int off = 128; off > 0; off >>= 1) {
        if ((int)threadIdx.x < off) {
            ss[threadIdx.x] += ss[threadIdx.x + off];
            sq[threadIdx.x] += sq[threadIdx.x + off];
        }
        __syncthreads();
    }
    if (threadIdx.x == 0) { xpart[blockIdx.x * 2] = ss[0]; xpart[blockIdx.x * 2 + 1] = sq[0]; }
}

// d = softplus(bn_d1(x*w1+b1)) == softplus(coefA*x + coefC), derived analytically
__global__ void k_dcoef(const float* __restrict__ xpart,
                        const float* __restrict__ w1, const float* __restrict__ b1,
                        const float* __restrict__ g1, const float* __restrict__ be1,
                        float* __restrict__ coefA, float* __restrict__ coefC) {
    __shared__ float mv[2];
    if (threadIdx.x == 0) {
        float s = 0.f, q = 0.f;
        for (int i = 0; i < XBLK; i++) { s += xpart[i * 2]; q += xpart[i * 2 + 1]; }
        float mx = s / (float)N_EDGES;
        mv[0] = mx;
        mv[1] = q / (float)N_EDGES - mx * mx;
    }
    __syncthreads();
    int j = threadIdx.x;
    if (j < HD) {
        float w = w1[j], b = b1[j];
        float mean  = w * mv[0] + b;
        float var   = w * w * mv[1];
        float scale = g1[j] * rsqrtf(var + BN_EPS);
        float shift = be1[j] - scale * mean;
        coefA[j] = scale * w;
        coefC[j] = scale * b + shift;
    }
}

// ---------------- edge GEMM: d2 = softplus_row(x) @ W2^T + b2, + BN partials --------
__global__ __launch_bounds__(256)
void k_edge_gemm(const float* __restrict__ x, const float* __restrict__ w2,
                 const float* __restrict__ b2, const float* __restrict__ cA,
                 const float* __restrict__ cC, float* __restrict__ d2,
                 float* __restrict__ part) {
    __shared__ float Bt[HD * HD];       // [k][o] = w2[o][k]
    __shared__ float Als[8][16 * 68];   // per-wave A tile, padded
    __shared__ float Xs[8][16];
    __shared__ float csum[HD], csq[HD];
    int tid = threadIdx.x;
    for (int i = tid; i < HD * HD; i += 256) {
        int k = i >> 6, o = i & 63;
        Bt[i] = w2[o * HD + k];
    }
    if (tid < HD) { csum[tid] = 0.f; csq[tid] = 0.f; }
    __syncthreads();

    int wave = tid >> 5, lane = tid & 31, l16 = lane & 15, half = lane >> 4;
    float* A = Als[wave];
    float* xs = Xs[wave];
    float sa[4] = {0, 0, 0, 0}, qa[4] = {0, 0, 0, 0};
    const int ntiles = N_EDGES / 16;
    const int tstride = gridDim.x * 8;
    for (int t = blockIdx.x * 8 + wave; t < ntiles; t += tstride) {
        int e0 = t * 16;
        if (lane < 16) xs[lane] = x[e0 + lane];
        if (t + tstride < ntiles)   // prefetch next tile's x into cache
            __builtin_prefetch(&x[(t + tstride) * 16], 0, 1);
        __builtin_amdgcn_wave_barrier(); asm volatile("" ::: "memory");
        for (int i = lane; i < 16 * HD; i += 32) {
            int r = i >> 6, k = i & 63;
            A[r * 68 + k] = softplus_f(cA[k] * xs[r] + cC[k]);
        }
        __builtin_amdgcn_wave_barrier(); asm volatile("" ::: "memory");
        #pragma unroll
        for (int nt = 0; nt < 4; nt++) {
            v8f c = vzero8();
            c = gemm_tile<64, 68, 64>(A, &Bt[nt * 16], l16, half, c);
            float bcol = b2[nt * 16 + l16];
            #pragma unroll
            for (int v = 0; v < 8; v++) {
                int m = v + 8 * half;
                float val = c[v] + bcol;
                d2[(size_t)(e0 + m) * HD + nt * 16 + l16] = val;
                sa[nt] += val; qa[nt] += val * val;
            }
        }
        asm volatile("" ::: "memory");
    }
    #pragma unroll
    for (int nt = 0; nt < 4; nt++) {
        atomicAdd(&csum[nt * 16 + l16], sa[nt]);
        atomicAdd(&csq [nt * 16 + l16], qa[nt]);
    }
    __syncthreads();
    if (tid < HD) {
        part[(size_t)blockIdx.x * 128 + tid]      = csum[tid];
        part[(size_t)blockIdx.x * 128 + 64 + tid] = csq[tid];
    }
}

// ---------------- BN finalize: partials -> per-column scale/shift ----------------
__global__ void k_bn_finalize(const float* __restrict__ part, int nb, int C, float cnt,
                              const float* __restrict__ g, const float* __restrict__ be,
                              float* __restrict__ scale, float* __restrict__ shift) {
    int j = threadIdx.x;
    if (j >= C) return;
    float s = 0.f, q = 0.f;
    for (int i = 0; i < nb; i++) {
        s += part[(size_t)i * 2 * C + j];
        q += part[(size_t)i * 2 * C + C + j];
    }
    float mean = s / cnt;
    float var  = q / cnt - mean * mean;
    float sc   = g[j] * rsqrtf(var + BN_EPS);
    scale[j] = sc;
    shift[j] = be[j] - sc * mean;
}

// ---------------- ea = bn_d2(d2) * edge_emb[edge_type]  (in place) ----------------
__global__ void k_ea(float* __restrict__ d2ea, const int* __restrict__ etype,
                     const float* __restrict__ edge_emb,
                     const float* __restrict__ scale, const float* __restrict__ shift) {
    int idx = blockIdx.x * blockDim.x + threadIdx.x;
    if (idx >= N_EDGES * 16) return;
    int e = idx >> 4, k0 = (idx & 15) * 4;
    float4 d = ((const float4*)d2ea)[idx];
    const float* em = &edge_emb[(size_t)etype[e] * HD + k0];
    float4 o;
    o.x = (scale[k0 + 0] * d.x + shift[k0 + 0]) * em[0];
    o.y = (scale[k0 + 1] * d.y + shift[k0 + 1]) * em[1];
    o.z = (scale[k0 + 2] * d.z + shift[k0 + 2]) * em[2];
    o.w = (scale[k0 + 3] * d.w + shift[k0 + 3]) * em[3];
    ((float4*)d2ea)[idx] = o;
}

// ---------------- node_attr gather (also seeds the segment-sum accumulator) -------
__global__ void k_gather_nodes(const int* __restrict__ ntype,
                               const float* __restrict__ node_emb,
                               float* __restrict__ h, float* __restrict__ acc) {
    int idx = blockIdx.x * blockDim.x + threadIdx.x;
    if (idx >= N_NODES * 16) return;
    int n = idx >> 4, qq = idx & 15;
    float4 v = ((const float4*)node_emb)[(size_t)ntype[n] * 16 + qq];
    ((float4*)h)[idx] = v;
    ((float4*)acc)[idx] = v;
}

// ---------------- gconv scatter: acc[dst] += softplus(h[src] + ea) ----------------
__global__ void k_gconv_scatter(const float* __restrict__ h, const float* __restrict__ ea,
                                const int* __restrict__ eidx, float* __restrict__ acc) {
    int idx = blockIdx.x * blockDim.x + threadIdx.x;
    if (idx >= N_EDGES * 16) return;
    int e = idx >> 4, qq = idx & 15;
    int src = eidx[e];
    int dst = eidx[N_EDGES + e];
    float4 hv = ((const float4*)h)[src * 16 + qq];
    float4 ev = ((const float4*)ea)[idx];
    float m0 = softplus_f(hv.x + ev.x);
    float m1 = softplus_f(hv.y + ev.y);
    float m2 = softplus_f(hv.z + ev.z);
    float m3 = softplus_f(hv.w + ev.w);
    float* dp = &acc[(size_t)dst * HD + qq * 4];
    atomicAdd(dp + 0, m0); atomicAdd(dp + 1, m1);
    atomicAdd(dp + 2, m2); atomicAdd(dp + 3, m3);
}

// ---------------- per-column partial stats over a [rows x C] matrix ----------------
__global__ __launch_bounds__(256)
void k_col_stats(const float* __restrict__ X, int rows, int C, float* __restrict__ part) {
    extern __shared__ float sm[];      // 2*C floats
    float* csum = sm; float* csq = sm + C;
    int C4 = C >> 2;
    int tid = threadIdx.x;
    for (int i = tid; i < 2 * C; i += blockDim.x) sm[i] = 0.f;
    __syncthreads();
    int total = rows * C4;
    int start = blockIdx.x * blockDim.x + tid;
    int col4 = start % C4;             // invariant: stride (grid*256) % C4 == 0
    float s[4] = {0, 0, 0, 0}, q[4] = {0, 0, 0, 0};
    for (int idx = start; idx < total; idx += gridDim.x * blockDim.x) {
        float4 v = ((const float4*)X)[idx];
        s[0] += v.x; q[0] += v.x * v.x;
        s[1] += v.y; q[1] += v.y * v.y;
        s[2] += v.z; q[2] += v.z * v.z;
        s[3] += v.w; q[3] += v.w * v.w;
    }
    #pragma unroll
    for (int i = 0; i < 4; i++) {
        atomicAdd(&csum[col4 * 4 + i], s[i]);
        atomicAdd(&csq [col4 * 4 + i], q[i]);
    }
    __syncthreads();
    for (int i = tid; i < C; i += blockDim.x) {
        part[(size_t)blockIdx.x * 2 * C + i]     = csum[i];
        part[(size_t)blockIdx.x * 2 * C + C + i] = csq[i];
    }
}

// ---------------- y = [softplus](scale*x + shift); optional second copy ------------
__global__ void k_bn_apply(const float* __restrict__ X, const float* __restrict__ scale,
                           const float* __restrict__ shift, int rows, int C, int do_sp,
                           float* __restrict__ Y1, float* __restrict__ Y2) {
    int C4 = C >> 2;
    int idx = blockIdx.x * blockDim.x + threadIdx.x;
    if (idx >= rows * C4) return;
    int k0 = (idx % C4) * 4;
    float4 v = ((const float4*)X)[idx];
    float4 o;
    o.x = scale[k0 + 0] * v.x + shift[k0 + 0];
    o.y = scale[k0 + 1] * v.y + shift[k0 + 1];
    o.z = scale[k0 + 2] * v.z + shift[k0 + 2];
    o.w = scale[k0 + 3] * v.w + shift[k0 + 3];
    if (do_sp) {
        o.x = softplus_f(o.x); o.y = softplus_f(o.y);
        o.z = softplus_f(o.z); o.w = softplus_f(o.w);
    }
    ((float4*)Y1)[idx] = o;
    if (Y2) ((float4*)Y2)[idx] = o;
}

__global__ void k_zero(float* __restrict__ p, int n) {
    int i = blockIdx.x * blockDim.x + threadIdx.x;
    if (i < n) p[i] = 0.f;
}

// ---------------- graph pooling: hg[batch[n]] += h[n] ----------------
__global__ void k_pool(const float* __restrict__ h, const int* __restrict__ batch,
                       float* __restrict__ hg) {
    int idx = blockIdx.x * blockDim.x + threadIdx.x;
    if (idx >= N_NODES * 16) return;
    int n = idx >> 4, qq = idx & 15;
    float4 v = ((const float4*)h)[idx];
    float* dp = &hg[(size_t)batch[n] * HD + qq * 4];
    atomicAdd(dp + 0, v.x); atomicAdd(dp + 1, v.y);
    atomicAdd(dp + 2, v.z); atomicAdd(dp + 3, v.w);
}

// ---------------- fc1: [h | hg[batch]] (N x 128) @ W^T -> N x 64, + BN partials ----
__global__ __launch_bounds__(128)
void k_fc1_gemm(const float* __restrict__ h, const float* __restrict__ hg,
                const int* __restrict__ batch,
                const float* __restrict__ w, const float* __restrict__ bias,
                float* __restrict__ out, float* __restrict__ part) {
    __shared__ float Bt[128 * 64];    // [k][o] = w[o][k], k<128
    __shared__ float Als[4][16 * 68]; // per-wave A tile for one 64-K stage
    __shared__ float csum[64], csq[64];
    int tid = threadIdx.x;
    for (int i = tid; i < 128 * 64; i += 128) {
        int k = i >> 6, o = i & 63;
        Bt[i] = w[o * 128 + k];
    }
    if (tid < 64) { csum[tid] = 0.f; csq[tid] = 0.f; }
    __syncthreads();

    int wave = tid >> 5, lane = tid & 31, l16 = lane & 15, half = lane >> 4;
    float* A = Als[wave];
    float sa[4] = {0, 0, 0, 0}, qa[4] = {0, 0, 0, 0};
    const int ntiles = N_NODES / 16;
    for (int t = blockIdx.x * 4 + wave; t < ntiles; t += gridDim.x * 4) {
        int nb0 = t * 16;
        // K stage 1: h rows (contiguous) via async global->LDS copy
        wait_ds0();                       // prior-tile WMMA LDS reads done (WAR)
        for (int i = lane; i < 16 * 16; i += 32) {   // 256 float4 per tile
            int r = i >> 4, q = i & 15;
            async_copy_b128(&A[r * 68 + q * 4], &h[(size_t)(nb0 + r) * 64 + q * 4]);
        }
        wait_async0();
        __builtin_amdgcn_wave_barrier(); asm volatile("" ::: "memory");
        v8f c[4];
        #pragma unroll
        for (int nt = 0; nt < 4; nt++) {
            c[nt] = vzero8();
            c[nt] = gemm_tile<64, 68, 64>(A, &Bt[nt * 16], l16, half, c[nt]);
        }
        asm volatile("" ::: "memory");
        // K stage 2: hg[batch[n]] rows (gathered) via async copy
        wait_ds0();                       // stage-1 WMMA LDS reads done (WAR)
        for (int i = lane; i < 16 * 16; i += 32) {
            int r = i >> 4, q = i & 15;
            int g = batch[nb0 + r];
            async_copy_b128(&A[r * 68 + q * 4], &hg[(size_t)g * 64 + q * 4]);
        }
        wait_async0();
        __builtin_amdgcn_wave_barrier(); asm volatile("" ::: "memory");
        #pragma unroll
        for (int nt = 0; nt < 4; nt++)
            c[nt] = gemm_tile<64, 68, 64>(A, &Bt[64 * 64 + nt * 16], l16, half, c[nt]);
        #pragma unroll
        for (int nt = 0; nt < 4; nt++) {
            float bcol = bias[nt * 16 + l16];
            #pragma unroll
            for (int v = 0; v < 8; v++) {
                int m = v + 8 * half;
                float val = c[nt][v] + bcol;
                out[(size_t)(nb0 + m) * 64 + nt * 16 + l16] = val;
                sa[nt] += val; qa[nt] += val * val;
            }
        }
        asm volatile("" ::: "memory");
    }
    #pragma unroll
    for (int nt = 0; nt < 4; nt++) {
        atomicAdd(&csum[nt * 16 + l16], sa[nt]);
        atomicAdd(&csq [nt * 16 + l16], qa[nt]);
    }
    __syncthreads();
    if (tid < 64) {
        part[(size_t)blockIdx.x * 128 + tid]      = csum[tid];
        part[(size_t)blockIdx.x * 128 + 64 + tid] = csq[tid];
    }
}

// ---------------- fc2: (N x 64) @ W^T -> N x 32, + BN partials ----------------
__global__ __launch_bounds__(256)
void k_fc2_gemm(const float* __restrict__ in, const float* __restrict__ w,
                const float* __restrict__ bias, float* __restrict__ out,
                float* __restrict__ part) {
    __shared__ float Bt[64 * 32];
    __shared__ float Als[8][16 * 68];
    __shared__ float csum[32], csq[32];
    int tid = threadIdx.x;
    for (int i = tid; i < 64 * 32; i += 256) {
        int k = i >> 5, o = i & 31;
        Bt[i] = w[o * 64 + k];
    }
    if (tid < 32) { csum[tid] = 0.f; csq[tid] = 0.f; }
    __syncthreads();
    int wave = tid >> 5, lane = tid & 31, l16 = lane & 15, half = lane >> 4;
    float* A = Als[wave];
    float sa[2] = {0, 0}, qa[2] = {0, 0};
    const int ntiles = N_NODES / 16;
    for (int t = blockIdx.x * 8 + wave; t < ntiles; t += gridDim.x * 8) {
        int nb0 = t * 16;
        wait_ds0();                       // prior-tile WMMA LDS reads done (WAR)
        for (int i = lane; i < 16 * 16; i += 32) {
            int r = i >> 4, q = i & 15;
            async_copy_b128(&A[r * 68 + q * 4], &in[(size_t)(nb0 + r) * 64 + q * 4]);
        }
        wait_async0();
        __builtin_amdgcn_wave_barrier(); asm volatile("" ::: "memory");
        #pragma unroll
        for (int nt = 0; nt < 2; nt++) {
            v8f c = vzero8();
            c = gemm_tile<64, 68, 32>(A, &Bt[nt * 16], l16, half, c);
            float bcol = bias[nt * 16 + l16];
            #pragma unroll
            for (int v = 0; v < 8; v++) {
                int m = v + 8 * half;
                float val = c[v] + bcol;
                out[(size_t)(nb0 + m) * 32 + nt * 16 + l16] = val;
                sa[nt] += val; qa[nt] += val * val;
            }
        }
        asm volatile("" ::: "memory");
    }
    atomicAdd(&csum[l16], sa[0]);      atomicAdd(&csq[l16], qa[0]);
    atomicAdd(&csum[16 + l16], sa[1]); atomicAdd(&csq[16 + l16], qa[1]);
    __syncthreads();
    if (tid < 32) {
        part[(size_t)blockIdx.x * 64 + tid]      = csum[tid];
        part[(size_t)blockIdx.x * 64 + 32 + tid] = csq[tid];
    }
}

// ---------------- fc3: (N x 32) @ W^T -> mu (N x 16) and sigma (N x 16) -----------
__global__ __launch_bounds__(256)
void k_fc3_gemm(const float* __restrict__ in, const float* __restrict__ w,
                const float* __restrict__ bias, float* __restrict__ outp) {
    __shared__ float Bt[32 * 32];
    __shared__ float Als[8][16 * 36];
    int tid = threadIdx.x;
    for (int i = tid; i < 32 * 32; i += 256) {
        int k = i >> 5, o = i & 31;
        Bt[i] = w[o * 32 + k];
    }
    __syncthreads();
    int wave = tid >> 5, lane = tid & 31, l16 = lane & 15, half = lane >> 4;
    float* A = Als[wave];
    const int ntiles = N_NODES / 16;
    for (int t = blockIdx.x * 8 + wave; t < ntiles; t += gridDim.x * 8) {
        int nb0 = t * 16;
        wait_ds0();                       // prior-tile WMMA LDS reads done (WAR)
        for (int i = lane; i < 16 * 8; i += 32) {   // 128 float4 per tile
            int r = i >> 3, q = i & 7;
            async_copy_b128(&A[r * 36 + q * 4], &in[(size_t)(nb0 + r) * 32 + q * 4]);
        }
        wait_async0();
        __builtin_amdgcn_wave_barrier(); asm volatile("" ::: "memory");
        #pragma unroll
        for (int nt = 0; nt < 2; nt++) {
            v8f c = vzero8();
            c = gemm_tile<32, 36, 32>(A, &Bt[nt * 16], l16, half, c);
            float bcol = bias[nt * 16 + l16];
            #pragma unroll
            for (int v = 0; v < 8; v++) {
                size_t row = (size_t)(nb0 + v + 8 * half);
                float val = c[v] + bcol;
                if (nt == 0) outp[row * 16 + l16] = val;                          // mu
                else         outp[(size_t)N_NODES * 16 + row * 16 + l16] = val;   // sigma
            }
        }
        asm volatile("" ::: "memory");
    }
}

// ---------------- host-side orchestration ----------------
extern "C" void kernel_launch(void* const* d_in, const int* in_sizes, int n_in,
                              void* d_out, int out_size, void* d_ws, size_t ws_size,
                              hipStream_t stream) {
    (void)in_sizes; (void)n_in; (void)out_size; (void)ws_size;
    const float* x        = (const float*)d_in[0];
    const int* node_type  = (const int*)d_in[1];
    const int* edge_type  = (const int*)d_in[2];
    const int* edge_index = (const int*)d_in[3];
    const int* batch      = (const int*)d_in[4];
    const float* node_emb = (const float*)d_in[5];
    const float* edge_emb = (const float*)d_in[6];
    const float* d1w = (const float*)d_in[7];
    const float* d1b = (const float*)d_in[8];
    const float* d2w = (const float*)d_in[9];
    const float* d2b = (const float*)d_in[10];
    const float* o1w = (const float*)d_in[11];
    const float* o1b = (const float*)d_in[12];
    const float* o2w = (const float*)d_in[13];
    const float* o2b = (const float*)d_in[14];
    const float* o3w = (const float*)d_in[15];
    const float* o3b = (const float*)d_in[16];
    const float* bn_d1_g = (const float*)d_in[17];
    const float* bn_d1_b = (const float*)d_in[18];
    const float* bn_d2_g = (const float*)d_in[19];
    const float* bn_d2_b = (const float*)d_in[20];
    const float* bn_c_g[3] = {(const float*)d_in[21], (const float*)d_in[23], (const float*)d_in[25]};
    const float* bn_c_b[3] = {(const float*)d_in[22], (const float*)d_in[24], (const float*)d_in[26]};
    const float* bn_o1_g = (const float*)d_in[27];
    const float* bn_o1_b = (const float*)d_in[28];
    const float* bn_o2_g = (const float*)d_in[29];
    const float* bn_o2_b = (const float*)d_in[30];

    float* ws = (float*)d_ws;
    size_t p = 0;
    float* EA    = ws + p; p += (size_t)N_EDGES * HD;   // d2, then ea (in place)
    float* HBUF  = ws + p; p += (size_t)N_NODES * HD;   // current h
    float* ACC   = ws + p; p += (size_t)N_NODES * HD;   // segment-sum acc / fc1 out
    float* HG    = ws + p; p += (size_t)N_GRAPH * HD;   // pooled per-graph
    float* PART  = ws + p; p += (size_t)NBLK * 128;     // BN partials
    float* SCALE = ws + p; p += 64;
    float* SHIFT = ws + p; p += 64;
    float* CA    = ws + p; p += 64;                     // d-layer affine coefs
    float* CC    = ws + p; p += 64;
    float* XPART = ws + p; p += 2 * XBLK;
    float* FC2   = EA;                                  // EA free after last gconv

    const int egrid = (N_EDGES * 16 + 255) / 256;       // 50000
    const int ngrid = (N_NODES * 16 + 255) / 256;       // 3125

    // d-layer: scalar x stats -> analytic BN_d1 coefficients
    k_xstats_partial<<<XBLK, 256, 0, stream>>>(x, XPART);
    k_dcoef<<<1, 64, 0, stream>>>(XPART, d1w, d1b, bn_d1_g, bn_d1_b, CA, CC);

    // d2 = d @ W2^T + b2 (WMMA), then ea = bn_d2(d2) * edge_emb[edge_type]
    k_edge_gemm<<<NBLK, 256, 0, stream>>>(x, d2w, d2b, CA, CC, EA, PART);
    k_bn_finalize<<<1, 64, 0, stream>>>(PART, NBLK, 64, (float)N_EDGES,
                                        bn_d2_g, bn_d2_b, SCALE, SHIFT);
    k_ea<<<egrid, 256, 0, stream>>>(EA, edge_type, edge_emb, SCALE, SHIFT);

    // node features + 3 graph convolutions
    k_gather_nodes<<<ngrid, 256, 0, stream>>>(node_type, node_emb, HBUF, ACC);
    for (int it = 0; it < 3; it++) {
        k_gconv_scatter<<<egrid, 256, 0, stream>>>(HBUF, EA, edge_index, ACC);
        k_col_stats<<<NBLK, 256, 2 * 64 * sizeof(float), stream>>>(ACC, N_NODES, 64, PART);
        k_bn_finalize<<<1, 64, 0, stream>>>(PART, NBLK, 64, (float)N_NODES,
                                            bn_c_g[it], bn_c_b[it], SCALE, SHIFT);
        int do_sp = (it < 2) ? 1 : 0;
        k_bn_apply<<<ngrid, 256, 0, stream>>>(ACC, SCALE, SHIFT, N_NODES, 64, do_sp,
                                              HBUF, do_sp ? ACC : nullptr);
    }

    // pooling + output head
    k_zero<<<(N_GRAPH * HD + 255) / 256, 256, 0, stream>>>(HG, N_GRAPH * HD);
    k_pool<<<ngrid, 256, 0, stream>>>(HBUF, batch, HG);

    k_fc1_gemm<<<NBLK, 128, 0, stream>>>(HBUF, HG, batch, o1w, o1b, ACC, PART);
    k_bn_finalize<<<1, 64, 0, stream>>>(PART, NBLK, 64, (float)N_NODES,
                                        bn_o1_g, bn_o1_b, SCALE, SHIFT);
    k_bn_apply<<<ngrid, 256, 0, stream>>>(ACC, SCALE, SHIFT, N_NODES, 64, 1, ACC, nullptr);

    k_fc2_gemm<<<NBLK, 256, 0, stream>>>(ACC, o2w, o2b, FC2, PART);
    k_bn_finalize<<<1, 32, 0, stream>>>(PART, NBLK, 32, (float)N_NODES,
                                        bn_o2_g, bn_o2_b, SCALE, SHIFT);
    int ngrid32 = (N_NODES * 8 + 255) / 256;
    k_bn_apply<<<ngrid32, 256, 0, stream>>>(FC2, SCALE, SHIFT, N_NODES, 32, 1, FC2, nullptr);

    k_fc3_gemm<<<NBLK, 256, 0, stream>>>(FC2, o3w, o3b, (float*)d_out);
}